// FlashKAN_56676388438745
// MI455X (gfx1250) — compile-verified
//
#include <hip/hip_runtime.h>
#include <math.h>

// FlashKAN forward on gfx1250 (MI455X), f32 WMMA path, round 3.
//
// y[b,o] = sum_i sum_g C_i[b,g] * w[g,i,o].  Dense-g expansion per channel
// -> 17 x V_WMMA_F32_16X16X4_F32 steps.  This round: BT=32 (two 16x16
// accumulators per wave, every B fragment reused twice -> w L2 traffic
// halved to ~285MB) and a g-contiguous LDS panel so each A fragment is one
// ds_load_b64.  Inner loop: 2 ds_load_b64 + 2 global_load_b32 + 2 wmma.

typedef __attribute__((ext_vector_type(2))) float v2f;
typedef __attribute__((ext_vector_type(4))) float v4f;
typedef __attribute__((ext_vector_type(8))) float v8f;

#define B_TOT  2048
#define I_DIM  128
#define O_DIM  128
#define NKNOT  71            // G + 2k + 1
#define BT     32            // b-rows per block (2 WMMA M-tiles)
#define CH     8             // channels per chunk
#define NCHUNK (I_DIM / CH)
#define GPAD   70            // g-stride pad (even for b64 align, bank spread)
#define PANEL_DW 18432       // 8*32*70 = 17920, padded to 256*4*18 for b128 fill

__global__ __launch_bounds__(256)
void flashkan_fwd(const float* __restrict__ x,
                  const float* __restrict__ w,
                  const float* __restrict__ t,
                  float* __restrict__ out)
{
    __shared__ __align__(16) float sPA[PANEL_DW];   // panel [CH][BT rows][GPAD]
    __shared__ float sT[NKNOT];

    const int tid = threadIdx.x;
    const int b0t = blockIdx.x * BT;

    if (tid < NKNOT) sT[tid] = t[tid];

    // WMMA fragment coordinates (wave32, 16x16x4 f32 layout)
    const int lane = tid & 31;
    const int wid  = tid >> 5;            // 8 waves -> 8 o-tiles (O = 128)
    const int o0   = wid * 16;
    const int M    = lane & 15;           // A row (within tile) / B column
    const int kb   = (lane >> 4) << 1;    // K sub-base: 0 or 2

    // Record-builder coordinates: 256 threads = CH(8) x BT(32) records/chunk
    const int chc = tid & 7;              // channel within chunk (fast -> x coalesced)
    const int Mc  = tid >> 3;             // b-row within tile (0..31)

    v8f accA = {0.f,0.f,0.f,0.f,0.f,0.f,0.f,0.f};   // rows b0t .. b0t+15
    v8f accB = {0.f,0.f,0.f,0.f,0.f,0.f,0.f,0.f};   // rows b0t+16 .. b0t+31

    for (int ic = 0; ic < NCHUNK; ++ic) {
        __syncthreads();                  // prior chunk's panel reads complete

        // ---- zero-fill panel: 18 x ds_store_b128 per thread ----
        #pragma unroll
        for (int j = 0; j < PANEL_DW / (256 * 4); ++j)
            ((v4f*)sPA)[tid + 256 * j] = (v4f){0.f, 0.f, 0.f, 0.f};

        // ---- de Boor record: one per thread per chunk ----
        const int i = ic * CH + chc;
        const float xv = x[(size_t)(b0t + Mc) * I_DIM + i];

        int cell = (int)floorf((xv + 1.0f) * 32.0f);      // t[3+c] = -1 + c/32
        cell = cell < 0 ? 0 : (cell > 63 ? 63 : cell);
        const int ik = cell + 3;                          // i_knot in [3,66]

        float Nv[4]; Nv[0] = 1.0f; Nv[1] = Nv[2] = Nv[3] = 0.0f;
        float left[3], right[3];
        #pragma unroll
        for (int j = 1; j <= 3; ++j) {
            left[j - 1]  = xv - sT[ik + 1 - j];
            right[j - 1] = sT[ik + j] - xv;
            float saved = 0.0f;
            #pragma unroll
            for (int r = 0; r < j; ++r) {
                const float temp = Nv[r] / (right[r] + left[j - 1 - r]);
                Nv[r] = saved + right[r] * temp;
                saved = left[j - 1 - r] * temp;
            }
            Nv[j] = saved;
        }
        const float sv = xv * (1.0f / (1.0f + __expf(-xv)));   // silu
        const int gb = ik - 3;                                  // band start g

        __syncthreads();                  // zero-fill visible before scatter

        // ---- scatter 5 nonzeros (band is g-contiguous in this layout) ----
        {
            const int base = (chc * BT + Mc) * GPAD;
            sPA[base + gb + 0] = Nv[0];
            sPA[base + gb + 1] = Nv[1];
            sPA[base + gb + 2] = Nv[2];
            sPA[base + gb + 3] = Nv[3];
            sPA[base + 67]     = sv;      // silu slot
        }

        __syncthreads();                  // panel ready

        // ---- WMMA: 8 channels x 17 K-steps, 2 M-tiles per step ----
        for (int ch = 0; ch < CH; ++ch) {
            const int ii = ic * CH + ch;
            const float* wi = w + (size_t)ii * O_DIM + o0 + M;  // w[g][ii][o0+M]
            __builtin_prefetch(wi + O_DIM, 0, 0);               // next channel's line
            const int a0base = (ch * BT + M) * GPAD;            // rows 0..15
            const int a1base = a0base + 16 * GPAD;              // rows 16..31

            #pragma unroll
            for (int kk = 0; kk < 17; ++kk) {                   // g = 4kk + kb + {0,1}
                const int gk = (kk << 2) + kb;

                const v2f A0 = *reinterpret_cast<const v2f*>(sPA + a0base + gk);
                const v2f A1 = *reinterpret_cast<const v2f*>(sPA + a1base + gk);
                const float b0 = wi[(size_t)gk * (I_DIM * O_DIM)];
                const float b1 = wi[(size_t)(gk + 1) * (I_DIM * O_DIM)];
                const v2f Bm = {b0, b1};

                accA = __builtin_amdgcn_wmma_f32_16x16x4_f32(
                    false, A0, false, Bm, (short)0, accA, false, false);
                accB = __builtin_amdgcn_wmma_f32_16x16x4_f32(
                    false, A1, false, Bm, (short)0, accB, false, false);
            }
        }
    }

    // ---- store two 16x16 f32 C/D tiles (VGPR r: M = r + 8*(lane>=16)) ----
    const int mh = (lane >> 4) << 3;
    const int n  = lane & 15;
    float* opA = out + (size_t)(b0t + mh) * O_DIM + o0 + n;
    float* opB = out + (size_t)(b0t + 16 + mh) * O_DIM + o0 + n;
    #pragma unroll
    for (int r = 0; r < 8; ++r) {
        opA[(size_t)r * O_DIM] = accA[r];
        opB[(size_t)r * O_DIM] = accB[r];
    }
}

extern "C" void kernel_launch(void* const* d_in, const int* in_sizes, int n_in,
                              void* d_out, int out_size, void* d_ws, size_t ws_size,
                              hipStream_t stream)
{
    const float* x = (const float*)d_in[0];   // [2048,128]
    const float* w = (const float*)d_in[1];   // [68,128,128]
    const float* t = (const float*)d_in[2];   // [71]
    float* out = (float*)d_out;               // [2048,128]

    dim3 grid(B_TOT / BT);                    // 64 blocks
    dim3 block(256);                          // 8 wave32s
    flashkan_fwd<<<grid, block, 0, stream>>>(x, w, t, out);
}